// GAPN_Critic_28802050686998
// MI455X (gfx1250) — compile-verified
//
#include <hip/hip_runtime.h>
#include <hip/hip_bf16.h>

// ---------------------------------------------------------------------------
// CDNA5 (gfx1250) implementation of a 3-layer GAT + critic MLP.
// Dense GEMMs use V_WMMA_F32_16X16X4_F32 (full fp32 precision; total GEMM
// work is ~1.6 GFLOP and nowhere near the bottleneck, so no downcast). The
// edge phase (L2-resident gathers + f32 atomics) dominates runtime.
// ---------------------------------------------------------------------------

typedef __attribute__((ext_vector_type(2))) float v2f;
typedef __attribute__((ext_vector_type(8))) float v8f;

// ------------------------------ WMMA GEMM ----------------------------------
// One wave32 per 16x16 output tile.  D = op(A[M,K]) * B[K,N] (+bias) (relu?)
// A fragment (32-bit A 16x4): lane<16 -> K=k0,k0+1 ; lane>=16 -> K=k0+2,k0+3
// B fragment (mirror):        same K split, N = lane&15
// C/D: VGPR i holds row i (lanes 0-15) / row i+8 (lanes 16-31)
// Straight-line inner loop: clamped-row A load + multiplicative lane mask
// (no EXEC manipulation -> EXEC stays all-1s through the WMMA, per ISA) and
// pointer-increment addressing (no per-iteration 64-bit multiplies).
template <bool RELU_IN>
__global__ __launch_bounds__(32)
void wmma_gemm_f32(const float* __restrict__ A, const float* __restrict__ B,
                   const float* __restrict__ bias, float* __restrict__ D,
                   int M, int N, int K, int relu_out)
{
    const int lane = threadIdx.x & 31;
    const int col  = lane & 15;
    const int half = lane >> 4;          // 0 or 1
    const int kb   = half << 1;          // 0 or 2
    const int rowA = blockIdx.x * 16 + col;
    const int colg = blockIdx.y * 16 + col;

    const int   rowAc = rowA < M ? rowA : (M - 1);   // always-valid address
    const float amask = rowA < M ? 1.0f : 0.0f;      // zero OOB rows

    const float* __restrict__ ap = A + (size_t)rowAc * K + kb;
    const float* __restrict__ bp = B + (size_t)kb * N + colg;
    const size_t bstep = (size_t)4 * N;

    v8f acc = {};
    for (int k0 = 0; k0 < K; k0 += 4) {
        v2f a;
        a.x = ap[0];
        a.y = ap[1];
        if (RELU_IN) { a.x = fmaxf(a.x, 0.0f); a.y = fmaxf(a.y, 0.0f); }
        a.x *= amask;
        a.y *= amask;
        v2f b;
        b.x = bp[0];
        b.y = bp[N];
        // 8-arg form: (neg_a, A, neg_b, B, c_mod, C, reuse_a, reuse_b)
        acc = __builtin_amdgcn_wmma_f32_16x16x4_f32(
                  false, a, false, b, (short)0, acc, false, false);
        ap += 4;
        bp += bstep;
    }

    const float bv = bias ? bias[colg] : 0.0f;
    #pragma unroll
    for (int i = 0; i < 8; ++i) {
        int row = blockIdx.x * 16 + i + half * 8;
        if (row < M) {
            float v = acc[i] + bv;
            if (relu_out) v = fmaxf(v, 0.0f);
            D[(size_t)row * N + colg] = v;
        }
    }
}

// --------------------------- per-node logits -------------------------------
// One wave32 per node: contiguous 256B row read (float2/lane) + shfl reduce.
__global__ __launch_bounds__(256)
void logits_kernel(const float* __restrict__ h,
                   const float* __restrict__ a_s,
                   const float* __restrict__ a_d,
                   float* __restrict__ ls, float* __restrict__ ld_,
                   int N, int H)
{
    int lane = threadIdx.x & 31;
    int n = blockIdx.x * (blockDim.x >> 5) + (threadIdx.x >> 5);
    if (n >= N) return;
    const float2* row = (const float2*)(h + (size_t)n * H);
    const float2* as2 = (const float2*)a_s;
    const float2* ad2 = (const float2*)a_d;
    float s = 0.0f, d = 0.0f;
    for (int f2 = lane; f2 < (H >> 1); f2 += 32) {
        float2 v = row[f2];
        float2 cs = as2[f2];
        float2 cd = ad2[f2];
        s = fmaf(v.x, cs.x, fmaf(v.y, cs.y, s));
        d = fmaf(v.x, cd.x, fmaf(v.y, cd.y, d));
    }
    for (int off = 16; off > 0; off >>= 1) {
        s += __shfl_xor(s, off, 32);
        d += __shfl_xor(d, off, 32);
    }
    if (lane == 0) { ls[n] = s; ld_[n] = d; }
}

// ------------------------ node-state init (m, denom) -----------------------
// m stored with order-preserving uint encoding; enc==0 acts as -inf sentinel
// (only ever decoded for nodes that received at least one edge).
__global__ void init_nodes_kernel(unsigned* __restrict__ m,
                                  float* __restrict__ denom, int N)
{
    int n = blockIdx.x * blockDim.x + threadIdx.x;
    if (n >= N) return;
    m[n] = 0u;
    denom[n] = 0.0f;
}

__device__ __forceinline__ unsigned enc_f32(float f) {
    unsigned u = __float_as_uint(f);
    return (u & 0x80000000u) ? ~u : (u | 0x80000000u);
}
__device__ __forceinline__ float dec_f32(unsigned e) {
    unsigned u = (e & 0x80000000u) ? (e & 0x7FFFFFFFu) : ~e;
    return __uint_as_float(u);
}

// ----------------- edge pass 1: logit + LeakyReLU + segment max ------------
__global__ void edge_logit_kernel(const int* __restrict__ src,
                                  const int* __restrict__ dst,
                                  const int* __restrict__ et,
                                  const float* __restrict__ ls,
                                  const float* __restrict__ ld_,
                                  const float* __restrict__ etb,
                                  float* __restrict__ e,
                                  unsigned* __restrict__ m, int E)
{
    int i = blockIdx.x * blockDim.x + threadIdx.x;
    if (i >= E) return;
    int d = dst[i];
    float v = ls[src[i]] + ld_[d] + etb[et[i]];
    v = (v > 0.0f) ? v : 0.2f * v;               // LeakyReLU(0.2)
    e[i] = v;
    atomicMax(&m[d], enc_f32(v));                // global_atomic_max_u32
}

// ----------------- edge pass 2: exp(e - max) + segment sum -----------------
__global__ void edge_exp_kernel(const int* __restrict__ dst,
                                const unsigned* __restrict__ m,
                                float* __restrict__ ew,
                                float* __restrict__ denom, int E)
{
    int i = blockIdx.x * blockDim.x + threadIdx.x;
    if (i >= E) return;
    int d = dst[i];
    float w = __expf(ew[i] - dec_f32(m[d]));
    ew[i] = w;
    atomicAdd(&denom[d], w);                     // global_atomic_add_f32
}

// ----------------- edge pass 3: weighted message scatter-add ---------------
// One wave32 per edge; each lane handles 2 features (H=64) via float2 loads.
__global__ __launch_bounds__(256)
void edge_scatter_kernel(const int* __restrict__ src,
                         const int* __restrict__ dst,
                         const float* __restrict__ w,
                         const float* __restrict__ denom,
                         const float* __restrict__ h,
                         float* __restrict__ out, int E, int H)
{
    int lane = threadIdx.x & 31;
    int eidx = blockIdx.x * (blockDim.x >> 5) + (threadIdx.x >> 5);
    if (eidx >= E) return;
    int s = src[eidx], d = dst[eidx];
    float coeff = w[eidx] / (denom[d] + 1e-16f);
    const float2* hrow = (const float2*)(h + (size_t)s * H);
    float* orow = out + (size_t)d * H;
    for (int f2 = lane; f2 < (H >> 1); f2 += 32) {
        float2 v = hrow[f2];
        atomicAdd(&orow[2 * f2 + 0], v.x * coeff);
        atomicAdd(&orow[2 * f2 + 1], v.y * coeff);
    }
}

// ---------------- global add pool with relu on node features ---------------
__global__ void pool_kernel(const float* __restrict__ x,
                            const int* __restrict__ batch,
                            float* __restrict__ g, int N, int H)
{
    int t = blockIdx.x * blockDim.x + threadIdx.x;
    int total = N * (H >> 1);
    if (t >= total) return;
    int n  = t / (H >> 1);
    int f2 = t - n * (H >> 1);
    float2 v = ((const float2*)(x + (size_t)n * H))[f2];
    v.x = fmaxf(v.x, 0.0f);
    v.y = fmaxf(v.y, 0.0f);
    int b = batch[n];
    atomicAdd(&g[(size_t)b * H + 2 * f2 + 0], v.x);
    atomicAdd(&g[(size_t)b * H + 2 * f2 + 1], v.y);
}

// ---------------- final projection [G,ENC] @ [ENC,1] + bias ----------------
__global__ __launch_bounds__(256)
void final_kernel(const float* __restrict__ g2, const float* __restrict__ W,
                  const float* __restrict__ b, float* __restrict__ out,
                  int G, int K)
{
    int lane = threadIdx.x & 31;
    int gi = blockIdx.x * (blockDim.x >> 5) + (threadIdx.x >> 5);
    if (gi >= G) return;
    float s = 0.0f;
    for (int f = lane; f < K; f += 32)
        s = fmaf(g2[(size_t)gi * K + f], W[f], s);
    for (int off = 16; off > 0; off >>= 1)
        s += __shfl_xor(s, off, 32);
    if (lane == 0) out[gi] = s + b[0];
}

// ---------------------------------------------------------------------------

static inline void launch_gemm(const float* A, const float* B, const float* bias,
                               float* D, int M, int N, int K,
                               bool relu_in, int relu_out, hipStream_t stream)
{
    dim3 grid((M + 15) / 16, N / 16);
    if (relu_in)
        wmma_gemm_f32<true><<<grid, dim3(32), 0, stream>>>(A, B, bias, D, M, N, K, relu_out);
    else
        wmma_gemm_f32<false><<<grid, dim3(32), 0, stream>>>(A, B, bias, D, M, N, K, relu_out);
}

extern "C" void kernel_launch(void* const* d_in, const int* in_sizes, int n_in,
                              void* d_out, int out_size, void* d_ws, size_t ws_size,
                              hipStream_t stream)
{
    (void)n_in; (void)ws_size;

    const float* x       = (const float*)d_in[0];
    const int*   src     = (const int*)d_in[1];
    const int*   dst     = (const int*)d_in[2];
    const int*   et      = (const int*)d_in[3];
    const int*   batch   = (const int*)d_in[4];
    // d_in[5] = num_graphs (device scalar; host-side G comes from out_size)
    const float* W[3]    = { (const float*)d_in[6], (const float*)d_in[7], (const float*)d_in[8] };
    const float* a_src   = (const float*)d_in[9];
    const float* a_dst   = (const float*)d_in[10];
    const float* et_bias = (const float*)d_in[11];
    const float* encW0   = (const float*)d_in[12];
    const float* encB0   = (const float*)d_in[13];
    const float* encW1   = (const float*)d_in[14];
    const float* encB1   = (const float*)d_in[15];
    const float* finW    = (const float*)d_in[16];
    const float* finB    = (const float*)d_in[17];

    const int H   = in_sizes[9] / 3;        // a_src is (3,H)
    const int F   = in_sizes[6] / H;        // W0 is (F,H)
    const int N   = in_sizes[0] / F;        // x is (N,F)
    const int E   = in_sizes[1];
    const int T   = in_sizes[11] / 3;       // et_bias is (3,T)
    const int ENC = in_sizes[13];           // enc_b0 is (ENC,)
    const int G   = out_size;               // output is (G,)

    // ---- workspace layout (all float-sized slots) ----
    float* base = (float*)d_ws;
    size_t off = 0;
    float*    xb0   = base + off; off += (size_t)N * H;
    float*    xb1   = base + off; off += (size_t)N * H;
    float*    hbuf  = base + off; off += (size_t)N * H;
    float*    ls    = base + off; off += (size_t)N;
    float*    ldt   = base + off; off += (size_t)N;
    unsigned* menc  = (unsigned*)(base + off); off += (size_t)N;
    float*    denom = base + off; off += (size_t)N;
    float*    ew    = base + off; off += (size_t)E;
    float*    ga    = base + off; off += (size_t)G * ENC;
    float*    gb    = base + off; off += (size_t)G * ENC;

    const int nblk256 = (N + 255) / 256;
    const int eblk256 = (E + 255) / 256;

    // ---- 3 GAT layers ----
    const float* in_ptr = x;
    int Kin = F;
    float* outs[3] = { xb0, xb1, xb0 };
    for (int i = 0; i < 3; ++i) {
        float* out = outs[i];
        // h = relu?(in) @ W_i   (relu folded into the A-operand load)
        launch_gemm(in_ptr, W[i], nullptr, hbuf, N, H, Kin, /*relu_in=*/(i > 0), 0, stream);
        // per-node attention logits (one wave per node)
        logits_kernel<<<(N + 7) / 8, 256, 0, stream>>>(hbuf, a_src + i * H, a_dst + i * H,
                                                       ls, ldt, N, H);
        // reset segment-max / denom / output accumulators
        init_nodes_kernel<<<nblk256, 256, 0, stream>>>(menc, denom, N);
        hipMemsetAsync(out, 0, sizeof(float) * (size_t)N * H, stream);
        // edge pipeline
        edge_logit_kernel<<<eblk256, 256, 0, stream>>>(src, dst, et, ls, ldt,
                                                       et_bias + i * T, ew, menc, E);
        edge_exp_kernel<<<eblk256, 256, 0, stream>>>(dst, menc, ew, denom, E);
        edge_scatter_kernel<<<(E + 7) / 8, 256, 0, stream>>>(src, dst, ew, denom,
                                                             hbuf, out, E, H);
        in_ptr = out;
        Kin = H;
    }

    // ---- global add pool (relu on node rows) ----
    hipMemsetAsync(ga, 0, sizeof(float) * (size_t)G * ENC, stream);
    {
        int total = N * (H >> 1);
        pool_kernel<<<(total + 255) / 256, 256, 0, stream>>>(in_ptr, batch, ga, N, H);
    }

    // ---- critic MLP (WMMA GEMMs with bias+relu epilogue) ----
    launch_gemm(ga, encW0, encB0, gb, G, ENC, H,   false, 1, stream);
    launch_gemm(gb, encW1, encB1, ga, G, ENC, ENC, false, 1, stream);
    final_kernel<<<(G + 7) / 8, 256, 0, stream>>>(ga, finW, finB, (float*)d_out, G, ENC);
}